// NeuralODE_10393820856347
// MI455X (gfx1250) — compile-verified
//
#include <hip/hip_runtime.h>
#include <math.h>

typedef __attribute__((ext_vector_type(2))) float v2f;
typedef __attribute__((ext_vector_type(8))) float v8f;

#define STATE 64
#define BATCH 32768
#define ROWS_PER_WG 64
#define THREADS 128
#define ACT 68           // padded LDS row stride (floats) to avoid bank conflicts

// LDS layout (float offsets)
#define OFF_W1   0
#define OFF_W2   4096
#define OFF_W3   8192
#define OFF_B1   12288
#define OFF_B2   12352
#define OFF_B3   12416
#define OFF_ACT  12480
#define ACT_TILE (ROWS_PER_WG * ACT)              // 4352 floats
#define SMEM_FLOATS (OFF_ACT + 8 * ACT_TILE)      // x, s, k1..k6

static __device__ __forceinline__ float fast_tanh(float x) {
#if __has_builtin(__builtin_amdgcn_tanhf)
  return __builtin_amdgcn_tanhf(x);
#elif __has_builtin(__builtin_amdgcn_tanh_f32)
  return __builtin_amdgcn_tanh_f32(x);
#else
  return tanhf(x);
#endif
}

static __device__ __forceinline__ float4 ld4s(const float* p) { return *(const float4*)p; }
static __device__ __forceinline__ void st4s(float* p, float4 v) { *(float4*)p = v; }

// One wave computes dst[16x64] = act( src[16x64] @ W[64x64] + bias ).
// B fragments (bw[kc*4+nt]) and bias scalars (bb[nt]) are register-resident.
static __device__ __forceinline__ void wave_mm(const float* src, const v2f* bw,
                                               const float* bb, float* dst,
                                               int m0, int ln, int hf, bool do_tanh) {
  v8f acc[4];
#pragma unroll
  for (int nt = 0; nt < 4; ++nt) {
    float bv = bb[nt];
    acc[nt] = (v8f){bv, bv, bv, bv, bv, bv, bv, bv};
  }
  const int row = m0 + ln;
#pragma unroll
  for (int kc = 0; kc < 16; ++kc) {
    v2f a = *(const v2f*)&src[row * ACT + kc * 4 + 2 * hf];
#pragma unroll
    for (int nt = 0; nt < 4; ++nt) {
      acc[nt] = __builtin_amdgcn_wmma_f32_16x16x4_f32(
          false, a, false, bw[kc * 4 + nt], (short)0, acc[nt], false, false);
    }
  }
#pragma unroll
  for (int nt = 0; nt < 4; ++nt) {
#pragma unroll
    for (int v = 0; v < 8; ++v) {
      float val = acc[nt][v];
      if (do_tanh) val = fast_tanh(val);
      dst[(m0 + v + 8 * hf) * ACT + nt * 16 + ln] = val;
    }
  }
}

__global__ __launch_bounds__(THREADS) void NeuralODE_dopri5_kernel(
    const float* __restrict__ x0, const float* __restrict__ W1, const float* __restrict__ b1,
    const float* __restrict__ W2, const float* __restrict__ b2,
    const float* __restrict__ W3, const float* __restrict__ b3, float* __restrict__ out) {
  extern __shared__ float smem[];
  const int tid = threadIdx.x;

  // ---- stage weights into LDS in B-fragment (K-pair) packed layout ----
  {
    const float* wg[3] = {W1, W2, W3};
    const int woff[3] = {OFF_W1, OFF_W2, OFF_W3};
#pragma unroll
    for (int m = 0; m < 3; ++m) {
      const float* Wg = wg[m];
      float* Wd = smem + woff[m];
      for (int idx = tid; idx < 4096; idx += THREADS) {
        int k = idx >> 6, n = idx & 63;
        Wd[(((k >> 1) << 6) + n) * 2 + (k & 1)] = Wg[idx];
      }
    }
    for (int idx = tid; idx < 192; idx += THREADS) {
      float v = (idx < 64) ? b1[idx] : (idx < 128) ? b2[idx - 64] : b3[idx - 128];
      smem[OFF_B1 + idx] = v;
    }
  }

  // ---- load x tile (64 rows) ----
  const int r0 = blockIdx.x * ROWS_PER_WG;
#pragma unroll
  for (int i = 0; i < 8; ++i) {
    int fid = tid + THREADS * i;
    int r = fid >> 4, c = (fid & 15) << 2;
    float4 v = ld4s(x0 + (size_t)(r0 + r) * STATE + c);
    st4s(smem + OFF_ACT + r * ACT + c, v);
  }
  __syncthreads();

  const int lane = tid & 31;
  const int wave = tid >> 5;
  const int ln = lane & 15;
  const int hf = lane >> 4;
  const int m0 = wave * 16;

  // ---- hoist all B fragments + biases into registers (loop-invariant) ----
  // 3 x 64 v2f = 384 VGPRs; LDS already limits us to 1 wave/SIMD so this is free.
  v2f bw[3][64];
  float bb[3][4];
  {
    const int woff[3] = {OFF_W1, OFF_W2, OFF_W3};
#pragma unroll
    for (int m = 0; m < 3; ++m) {
      const float* Wp = smem + woff[m];
#pragma unroll
      for (int kc = 0; kc < 16; ++kc) {
#pragma unroll
        for (int nt = 0; nt < 4; ++nt) {
          bw[m][kc * 4 + nt] = *(const v2f*)&Wp[((2 * kc + hf) * 64 + nt * 16 + ln) * 2];
        }
      }
#pragma unroll
      for (int nt = 0; nt < 4; ++nt) bb[m][nt] = smem[OFF_B1 + m * 64 + nt * 16 + ln];
    }
  }

  float* xbuf = smem + OFF_ACT;
  float* sbuf = smem + OFF_ACT + ACT_TILE;
  float* kbuf[6];
#pragma unroll
  for (int j = 0; j < 6; ++j) kbuf[j] = smem + OFF_ACT + ACT_TILE * (2 + j);

  const float hstep = 1.0f / 16.0f;
  constexpr float AC[5][5] = {
      {1.0f / 5, 0, 0, 0, 0},
      {3.0f / 40, 9.0f / 40, 0, 0, 0},
      {44.0f / 45, -56.0f / 15, 32.0f / 9, 0, 0},
      {19372.0f / 6561, -25360.0f / 2187, 64448.0f / 6561, -212.0f / 729, 0},
      {9017.0f / 3168, -355.0f / 33, 46732.0f / 5247, 49.0f / 176, -5103.0f / 18656}};
  constexpr float BC[6] = {35.0f / 384, 0.0f, 500.0f / 1113, 125.0f / 192,
                           -2187.0f / 6784, 11.0f / 84};

#pragma unroll 1
  for (int step = 0; step < 16; ++step) {
#pragma unroll
    for (int st = 0; st < 6; ++st) {
      const float* src;
      if (st == 0) {
        src = xbuf;
      } else {
        // s = x + h * sum_{j<st} AC[st-1][j] * k_j  (this wave's 16 rows only)
#pragma unroll
        for (int i = 0; i < 8; ++i) {
          int r = m0 + hf + 2 * i;
          int c = ln << 2;
          float4 v = ld4s(xbuf + r * ACT + c);
#pragma unroll
          for (int j = 0; j < st; ++j) {
            float cf = hstep * AC[st - 1][j];
            float4 kv = ld4s(kbuf[j] + r * ACT + c);
            v.x = fmaf(cf, kv.x, v.x);
            v.y = fmaf(cf, kv.y, v.y);
            v.z = fmaf(cf, kv.z, v.z);
            v.w = fmaf(cf, kv.w, v.w);
          }
          st4s(sbuf + r * ACT + c, v);
        }
        src = sbuf;
      }
      // k_st = f(src): h1 -> kbuf[st], h2 -> sbuf, k -> kbuf[st]
      wave_mm(src,      bw[0], bb[0], kbuf[st], m0, ln, hf, true);
      wave_mm(kbuf[st], bw[1], bb[1], sbuf,     m0, ln, hf, true);
      wave_mm(sbuf,     bw[2], bb[2], kbuf[st], m0, ln, hf, false);
    }
    // x += h * (b1 k1 + b3 k3 + b4 k4 + b5 k5 + b6 k6)
#pragma unroll
    for (int i = 0; i < 8; ++i) {
      int r = m0 + hf + 2 * i;
      int c = ln << 2;
      float4 v = ld4s(xbuf + r * ACT + c);
#pragma unroll
      for (int j = 0; j < 6; ++j) {
        if (j == 1) continue;  // BC[1] == 0
        float cf = hstep * BC[j];
        float4 kv = ld4s(kbuf[j] + r * ACT + c);
        v.x = fmaf(cf, kv.x, v.x);
        v.y = fmaf(cf, kv.y, v.y);
        v.z = fmaf(cf, kv.z, v.z);
        v.w = fmaf(cf, kv.w, v.w);
      }
      st4s(xbuf + r * ACT + c, v);
    }
  }

  // ---- write back this wave's 16 rows ----
#pragma unroll
  for (int i = 0; i < 8; ++i) {
    int r = m0 + hf + 2 * i;
    int c = ln << 2;
    float4 v = ld4s(xbuf + r * ACT + c);
    st4s(out + (size_t)(r0 + r) * STATE + c, v);
  }
}

extern "C" void kernel_launch(void* const* d_in, const int* in_sizes, int n_in,
                              void* d_out, int out_size, void* d_ws, size_t ws_size,
                              hipStream_t stream) {
  const float* x0 = (const float*)d_in[0];
  const float* W1 = (const float*)d_in[1];
  const float* b1 = (const float*)d_in[2];
  const float* W2 = (const float*)d_in[3];
  const float* b2 = (const float*)d_in[4];
  const float* W3 = (const float*)d_in[5];
  const float* b3 = (const float*)d_in[6];
  float* out = (float*)d_out;

  dim3 grid(BATCH / ROWS_PER_WG);
  dim3 block(THREADS);
  size_t smem_bytes = (size_t)SMEM_FLOATS * sizeof(float);  // ~185 KB, fits 320 KB WGP LDS
  hipLaunchKernelGGL(NeuralODE_dopri5_kernel, grid, block, smem_bytes, stream,
                     x0, W1, b1, W2, b2, W3, b3, out);
}